// GAT_61057255080325
// MI455X (gfx1250) — compile-verified
//
#include <hip/hip_runtime.h>
#include <hip/hip_bf16.h>
#include <math.h>

// ---------------- problem constants (match reference) ----------------
constexpr int cN0 = 100000, cN1 = 50000, cN2 = 25000;
constexpr int cE0 = 400000, cE1 = 200000;
constexpr int cH1 = 4, cD1 = 64;   // layer0: 4 heads x 64
constexpr int cH2 = 1, cD2 = 64;   // layer1: 1 head  x 64
constexpr float cSLOPE = 0.2f;

typedef __attribute__((ext_vector_type(2))) float v2f;
typedef __attribute__((ext_vector_type(4))) float v4f;
typedef __attribute__((ext_vector_type(8))) float v8f;

// ---------------- fp32 WMMA GEMM: C = A(MxK) * B(KxN), K == KK ----------------
// Block = 8 waves; block computes a (16*MT) x (16*NT) strip of C.
// B panel (KK x 16*NT) is staged once in LDS as {B[k][c],B[k+1][c]} pairs so
// each WMMA B operand is a single ds_load_b64.
// Operand layouts for V_WMMA_F32_16X16X4_F32:
//   A 16x4: lanes 0-15 row=lane {K0,K1}; lanes 16-31 row=lane-16 {K2,K3}.
//   B 4x16: lanes 0-15 col=lane {K0,K1}; lanes 16-31 col=lane-16 {K2,K3}.
//   C/D:    VGPR i -> (row = i + 8*half, col = lane&15).
template <int MT, int NT, int KK>
__global__ __launch_bounds__(32 * MT) void wmma_gemm_f32(
    const float* __restrict__ A, const float* __restrict__ B,
    float* __restrict__ C, int M, int N) {
  constexpr int NC = 16 * NT;              // cols per block (64)
  constexpr int NB = (KK / 4) * 2 * NC;    // LDS float2 entries (8192)
  __shared__ v2f Bl[NB];                   // 64 KB

  const int ngroups = N / NC;
  const int ng = blockIdx.x % ngroups;
  const int mblock = blockIdx.x / ngroups;
  const int c0 = ng * NC;

  // ---- cooperative B panel load (coalesced over columns) ----
  const float* Bp = B + c0;
  for (int idx = threadIdx.x; idx < NB; idx += 32 * MT) {
    const int col = idx & (NC - 1);
    const int kh = idx / NC;               // k4*2 + half
    const int krow = (kh >> 1) * 4 + (kh & 1) * 2;
    v2f v;
    v.x = Bp[(size_t)krow * N + col];
    v.y = Bp[(size_t)(krow + 1) * N + col];
    Bl[idx] = v;
  }
  __syncthreads();

  const int waveid = threadIdx.x >> 5;
  const int lane = threadIdx.x & 31;
  const int mtile = mblock * MT + waveid;
  if (mtile * 16 >= M) return;             // no barriers past this point

  const int half = lane >> 4;              // 0: K0/K1 side, 1: K2/K3 side
  const int l = lane & 15;
  const float* Arow = A + (size_t)(mtile * 16 + l) * KK;

  v8f acc[NT] = {};

  for (int kc = 0; kc < KK; kc += 16) {
    // batch-fetch A (4 x b64 global) and B (16 x b64 LDS) for this chunk
    v2f a[4];
#pragma unroll
    for (int j = 0; j < 4; ++j) {
      const int ka = kc + j * 4 + half * 2;
      a[j].x = Arow[ka];
      a[j].y = Arow[ka + 1];
    }
    v2f b[4][NT];
#pragma unroll
    for (int j = 0; j < 4; ++j) {
      const int base = ((kc / 4 + j) * 2 + half) * NC + l;
#pragma unroll
      for (int t = 0; t < NT; ++t) b[j][t] = Bl[base + t * 16];
    }
#pragma unroll
    for (int j = 0; j < 4; ++j) {
#pragma unroll
      for (int t = 0; t < NT; ++t) {
        acc[t] = __builtin_amdgcn_wmma_f32_16x16x4_f32(
            false, a[j], false, b[j][t], (short)0, acc[t], false, false);
      }
    }
  }

  const int row0 = mtile * 16 + half * 8;
#pragma unroll
  for (int t = 0; t < NT; ++t) {
    const int col = c0 + t * 16 + l;
#pragma unroll
    for (int i = 0; i < 8; ++i) {
      C[(size_t)(row0 + i) * N + col] = acc[t][i];
    }
  }
}

// ---------------- helpers ----------------
__device__ inline void atomicMaxFloat(float* addr, float value) {
  // sign-split trick -> native integer atomics
  if (value >= 0.0f)
    atomicMax((int*)addr, __float_as_int(value));
  else
    atomicMin((unsigned int*)addr, __float_as_uint(value));
}

__global__ __launch_bounds__(256) void fill_f32(float* __restrict__ p, float v, int n) {
  int t = blockIdx.x * blockDim.x + threadIdx.x;
  if (t < n) p[t] = v;
}

// el[n,h] = <f[n,h,:], al[h,:]> (all src rows); er likewise for first Ndst rows
__global__ __launch_bounds__(256) void attn_dots(
    const float* __restrict__ f, const float* __restrict__ al,
    const float* __restrict__ ar, float* __restrict__ el,
    float* __restrict__ er, int Nsrc, int Ndst, int H, int D) {
  int t = blockIdx.x * blockDim.x + threadIdx.x;
  if (t >= Nsrc * H) return;
  int n = t / H, h = t - n * H;
  const v4f* fr = (const v4f*)(f + (size_t)n * H * D + (size_t)h * D);
  const v4f* alh = (const v4f*)(al + h * D);
  const v4f* arh = (const v4f*)(ar + h * D);
  float sl = 0.f, sr = 0.f;
#pragma unroll 4
  for (int d = 0; d < D / 4; ++d) {
    v4f v = fr[d], a = alh[d], b = arh[d];
    sl += v.x * a.x + v.y * a.y + v.z * a.z + v.w * a.w;
    sr += v.x * b.x + v.y * b.y + v.z * b.z + v.w * b.w;
  }
  el[t] = sl;
  if (n < Ndst) er[t] = sr;
}

// e = leaky_relu(el[src]+er[dst]); m[dst] = max(e) (atomic segment max)
__global__ __launch_bounds__(256) void edge_scores(
    const int* __restrict__ src, const int* __restrict__ dst,
    const float* __restrict__ el, const float* __restrict__ er,
    float* __restrict__ e, float* __restrict__ m, int E, int H) {
  int t = blockIdx.x * blockDim.x + threadIdx.x;
  if (t >= E * H) return;
  int edge = t / H, h = t - edge * H;
  int sN = src[edge], dN = dst[edge];
  float v = el[sN * H + h] + er[dN * H + h];
  v = v > 0.f ? v : cSLOPE * v;
  e[t] = v;
  atomicMaxFloat(&m[dN * H + h], v);
}

// e <- exp(e - m[dst]) in place; s[dst] += e (atomic segment sum)
__global__ __launch_bounds__(256) void edge_exp_sum(
    const int* __restrict__ dst, float* __restrict__ e,
    const float* __restrict__ m, float* __restrict__ s, int E, int H) {
  int t = blockIdx.x * blockDim.x + threadIdx.x;
  if (t >= E * H) return;
  int edge = t / H, h = t - edge * H;
  int dN = dst[edge];
  float v = __expf(e[t] - m[dN * H + h]);
  e[t] = v;
  unsafeAtomicAdd(&s[dN * H + h], v);
}

// out[dst, h, d] += f[src, h, d] * (ee / s[dst, h])
// one thread per 4 edge-features: b128 gather + 4 native fp32 atomics
__global__ __launch_bounds__(256) void edge_aggregate4(
    const int* __restrict__ src, const int* __restrict__ dst,
    const float* __restrict__ f, const float* __restrict__ ee,
    const float* __restrict__ ssum, float* __restrict__ out,
    int E, int H, int D) {
  const int C4 = (H * D) / 4;              // vec4 features per edge
  int t = blockIdx.x * blockDim.x + threadIdx.x;
  if (t >= E * C4) return;
  int edge = t / C4;
  int c4 = t - edge * C4;
  int h = (c4 * 4) / D;
  int sN = src[edge], dN = dst[edge];
  float alpha = ee[edge * H + h] / ssum[dN * H + h];
  const v4f v = *(const v4f*)(f + (size_t)sN * (H * D) + (size_t)c4 * 4);
  float* o = out + (size_t)dN * (H * D) + (size_t)c4 * 4;
  unsafeAtomicAdd(o + 0, v.x * alpha);
  unsafeAtomicAdd(o + 1, v.y * alpha);
  unsafeAtomicAdd(o + 2, v.z * alpha);
  unsafeAtomicAdd(o + 3, v.w * alpha);
}

// out[n,c] = act(in[n,c] + bias[c]); out may alias in
__global__ __launch_bounds__(256) void bias_act(
    const float* __restrict__ in, const float* __restrict__ bias,
    float* __restrict__ out, int Nn, int C, int do_relu) {
  int t = blockIdx.x * blockDim.x + threadIdx.x;
  if (t >= Nn * C) return;
  int c = t % C;
  float v = in[t] + bias[c];
  if (do_relu) v = fmaxf(v, 0.f);
  out[t] = v;
}

static inline int cdiv(long long a, int b) { return (int)((a + b - 1) / b); }

extern "C" void kernel_launch(void* const* d_in, const int* in_sizes, int n_in,
                              void* d_out, int out_size, void* d_ws, size_t ws_size,
                              hipStream_t stream) {
  (void)in_sizes; (void)n_in; (void)out_size; (void)ws_size;
  const float* x   = (const float*)d_in[0];
  const int*   e0s = (const int*)d_in[1];
  const int*   e0d = (const int*)d_in[2];
  const int*   e1s = (const int*)d_in[3];
  const int*   e1d = (const int*)d_in[4];
  const float* W1  = (const float*)d_in[5];
  const float* al1 = (const float*)d_in[6];
  const float* ar1 = (const float*)d_in[7];
  const float* b1  = (const float*)d_in[8];
  const float* W2  = (const float*)d_in[9];
  const float* al2 = (const float*)d_in[10];
  const float* ar2 = (const float*)d_in[11];
  const float* b2  = (const float*)d_in[12];
  float* out = (float*)d_out;

  // ---- workspace layout (floats) ----
  float* ws = (float*)d_ws;
  size_t off = 0;
  auto alloc = [&](size_t n) { float* p = ws + off; off += n; return p; };
  const int F1 = cH1 * cD1;                 // 256
  const int F2 = cH2 * cD2;                 // 64
  float* f1   = alloc((size_t)cN0 * F1);    // projected feats layer0
  float* el1  = alloc((size_t)cN0 * cH1);
  float* er1  = alloc((size_t)cN1 * cH1);
  float* m1   = alloc((size_t)cN1 * cH1);
  float* s1   = alloc((size_t)cN1 * cH1);
  float* e1b  = alloc((size_t)cE0 * cH1);   // e, then ee (in place)
  float* acc1 = alloc((size_t)cN1 * F1);    // layer0 aggregate -> h1 (in place)
  float* f2   = alloc((size_t)cN1 * F2);
  float* el2  = alloc((size_t)cN1 * cH2);
  float* er2  = alloc((size_t)cN2 * cH2);
  float* m2   = alloc((size_t)cN2 * cH2);
  float* s2   = alloc((size_t)cN2 * cH2);
  float* e2b  = alloc((size_t)cE1 * cH2);
  float* acc2 = alloc((size_t)cN2 * F2);

  const float NEG_INF = -INFINITY;
  constexpr int MT = 8, NT = 4, KK = 256;

  // ================= layer 0 =================
  {
    // GEMM: f1 = x(100000x256) @ W1(256x256)
    const int mblocks = cdiv(cN0 / 16, MT);
    const int ngroups = F1 / (16 * NT);
    wmma_gemm_f32<MT, NT, KK><<<mblocks * ngroups, 32 * MT, 0, stream>>>(
        x, W1, f1, cN0, F1);

    attn_dots<<<cdiv((long long)cN0 * cH1, 256), 256, 0, stream>>>(
        f1, al1, ar1, el1, er1, cN0, cN1, cH1, cD1);

    fill_f32<<<cdiv((long long)cN1 * cH1, 256), 256, 0, stream>>>(m1, NEG_INF, cN1 * cH1);
    fill_f32<<<cdiv((long long)cN1 * cH1, 256), 256, 0, stream>>>(s1, 0.f, cN1 * cH1);
    fill_f32<<<cdiv((long long)cN1 * F1, 256), 256, 0, stream>>>(acc1, 0.f, cN1 * F1);

    edge_scores<<<cdiv((long long)cE0 * cH1, 256), 256, 0, stream>>>(
        e0s, e0d, el1, er1, e1b, m1, cE0, cH1);
    edge_exp_sum<<<cdiv((long long)cE0 * cH1, 256), 256, 0, stream>>>(
        e0d, e1b, m1, s1, cE0, cH1);
    edge_aggregate4<<<cdiv((long long)cE0 * (F1 / 4), 256), 256, 0, stream>>>(
        e0s, e0d, f1, e1b, s1, acc1, cE0, cH1, cD1);

    // h1 = relu(acc1 + b1), in place
    bias_act<<<cdiv((long long)cN1 * F1, 256), 256, 0, stream>>>(
        acc1, b1, acc1, cN1, F1, 1);
  }

  // ================= layer 1 =================
  {
    // GEMM: f2 = h1(50000x256) @ W2(256x64)
    const int mblocks = cdiv(cN1 / 16, MT);
    const int ngroups = F2 / (16 * NT);
    wmma_gemm_f32<MT, NT, KK><<<mblocks * ngroups, 32 * MT, 0, stream>>>(
        acc1, W2, f2, cN1, F2);

    attn_dots<<<cdiv((long long)cN1 * cH2, 256), 256, 0, stream>>>(
        f2, al2, ar2, el2, er2, cN1, cN2, cH2, cD2);

    fill_f32<<<cdiv((long long)cN2 * cH2, 256), 256, 0, stream>>>(m2, NEG_INF, cN2 * cH2);
    fill_f32<<<cdiv((long long)cN2 * cH2, 256), 256, 0, stream>>>(s2, 0.f, cN2 * cH2);
    fill_f32<<<cdiv((long long)cN2 * F2, 256), 256, 0, stream>>>(acc2, 0.f, cN2 * F2);

    edge_scores<<<cdiv((long long)cE1 * cH2, 256), 256, 0, stream>>>(
        e1s, e1d, el2, er2, e2b, m2, cE1, cH2);
    edge_exp_sum<<<cdiv((long long)cE1 * cH2, 256), 256, 0, stream>>>(
        e1d, e2b, m2, s2, cE1, cH2);
    edge_aggregate4<<<cdiv((long long)cE1 * (F2 / 4), 256), 256, 0, stream>>>(
        e1s, e1d, f2, e2b, s2, acc2, cE1, cH2, cD2);

    // final: out = acc2 + b2 (no activation)
    bias_act<<<cdiv((long long)cN2 * F2, 256), 256, 0, stream>>>(
        acc2, b2, out, cN2, F2, 0);
  }
}